// QueryLoss_38302518346315
// MI455X (gfx1250) — compile-verified
//
#include <hip/hip_runtime.h>
#include <hip/hip_bf16.h>

#define IGNORE_INDEX (-100)

typedef float v2f __attribute__((ext_vector_type(2)));
typedef float v4f __attribute__((ext_vector_type(4)));
typedef float v8f __attribute__((ext_vector_type(8)));
typedef int   v4i __attribute__((ext_vector_type(4)));

// ---------------------------------------------------------------------------
// gfx1250 async global->LDS copy (ASYNCcnt path), with synchronous fallback.
// Builtin prototype (from hipcc diagnostic): (int4 AS1* src, int4 AS3* dst,
// imm offset, imm cpol).
// ---------------------------------------------------------------------------
#if __has_builtin(__builtin_amdgcn_global_load_async_to_lds_b128)
#define HAS_ASYNC_LDS 1
typedef __attribute__((address_space(1))) v4i g_v4i;
typedef __attribute__((address_space(3))) v4i l_v4i;
__device__ __forceinline__ void async_cp16(void* lds_dst, const void* gsrc) {
    __builtin_amdgcn_global_load_async_to_lds_b128((g_v4i*)gsrc, (l_v4i*)lds_dst,
                                                   0 /*offset*/, 0 /*cpol*/);
}
__device__ __forceinline__ void wait_async0() {
#if __has_builtin(__builtin_amdgcn_s_wait_asynccnt)
    __builtin_amdgcn_s_wait_asynccnt(0);
#else
    asm volatile("s_wait_asynccnt 0x0" ::: "memory");
#endif
}
#else
#define HAS_ASYNC_LDS 0
__device__ __forceinline__ void async_cp16(void* lds_dst, const void* gsrc) {
    *(v4f*)lds_dst = *(const v4f*)gsrc;     // VGPR round-trip fallback
}
__device__ __forceinline__ void wait_async0() {}
#endif

// ---------------------------------------------------------------------------
// Kernel 1: per-row squared norms of query  (Q rows x 512)
// ---------------------------------------------------------------------------
__global__ void qsqnorm_kernel(const float* __restrict__ q, float* __restrict__ sqn) {
    __shared__ float red[128];
    const int row = blockIdx.x;
    const float* qr = q + (size_t)row * 512;
    v4f v = *(const v4f*)(qr + threadIdx.x * 4);
    red[threadIdx.x] = v.x * v.x + v.y * v.y + v.z * v.z + v.w * v.w;
    __syncthreads();
    for (int off = 64; off > 0; off >>= 1) {
        if (threadIdx.x < off) red[threadIdx.x] += red[threadIdx.x + off];
        __syncthreads();
    }
    if (threadIdx.x == 0) sqn[row] = red[0];
}

// ---------------------------------------------------------------------------
// Kernel 2: per-row cross-entropy NLL, single-pass online softmax.
// One exp per element (branchy rescale) keeps this HBM-bound, not VALU-bound.
// ---------------------------------------------------------------------------
__global__ void ce_rows_kernel(const float* __restrict__ logits,
                               const int* __restrict__ label,
                               float* __restrict__ nll, int V) {
    __shared__ float smax[256];
    __shared__ float ssum[256];
    const int row = blockIdx.x;
    const float* lr = logits + (size_t)row * V;
    const v4f* lr4 = (const v4f*)lr;
    const int V4 = V >> 2;                       // V = 32000 divisible by 4

    float m = -INFINITY, s = 0.0f;
    for (int j = threadIdx.x; j < V4; j += 256) {
        v4f x = lr4[j];
#pragma unroll
        for (int c = 0; c < 4; ++c) {
            float xv = x[c];
            if (xv > m) {                        // rare after warm-up
                s = s * __expf(m - xv) + 1.0f;
                m = xv;
            } else {
                s += __expf(xv - m);
            }
        }
    }
    smax[threadIdx.x] = m;
    ssum[threadIdx.x] = s;
    __syncthreads();
    for (int off = 128; off > 0; off >>= 1) {
        if (threadIdx.x < off) {
            float ma = smax[threadIdx.x], mb = smax[threadIdx.x + off];
            float sa = ssum[threadIdx.x], sb = ssum[threadIdx.x + off];
            float M = fmaxf(ma, mb);
            smax[threadIdx.x] = M;
            ssum[threadIdx.x] = sa * expf(ma - M) + sb * expf(mb - M);
        }
        __syncthreads();
    }
    if (threadIdx.x == 0) {
        int l = label[row];
        float r = 0.0f;
        if (l != IGNORE_INDEX) {
            int cl = l < 0 ? 0 : (l > V - 1 ? V - 1 : l);
            r = (smax[0] + logf(ssum[0])) - lr[cl];
        }
        nll[row] = r;
    }
}

// ---------------------------------------------------------------------------
// Kernel 3: fused symmetric Gram + penalty using V_WMMA_F32_16X16X4_F32.
// 128x128 block tile, 8 wave32s, each wave owns 32x64 (8 accumulators).
// Double-buffered LDS; chunk k+1 copied via async global->LDS under the
// 64 WMMAs of chunk k. Symmetry: bx<=by only, off-diagonal weighted x2.
// ---------------------------------------------------------------------------
#define BT  128  // block tile
#define KC  32   // K chunk staged in LDS
#define LDP 36   // padded row stride: 144B = 16B-aligned rows, perfect banks

__global__ void gram_penalty_kernel(const float* __restrict__ q,
                                    const float* __restrict__ sqn,
                                    float* __restrict__ partial) {
    const int bx = blockIdx.x;   // row-block
    const int by = blockIdx.y;   // col-block
    const int pid = bx * gridDim.y + by;

    if (bx > by) {               // mirror handled by weight-2 on (bx,by)
        if (threadIdx.x == 0) partial[pid] = 0.0f;
        return;
    }

    __shared__ float As[2][BT][LDP];
    __shared__ float Bs[2][BT][LDP];
    __shared__ float red[256];

    const int tid  = threadIdx.x;
    const int w    = tid >> 5;       // wave id 0..7
    const int lane = tid & 31;
    const int half = lane >> 4;      // 0: lanes 0-15, 1: lanes 16-31
    const int l16  = lane & 15;

    const int R0 = bx * BT;
    const int C0 = by * BT;
    const int rowOff = (w >> 1) * 32;   // 0,32,64,96
    const int colOff = (w & 1) * 64;    // 0 or 64

    // per-thread staging coordinates: 1024 16B copies per operand, 4/thread
    const int r0g = tid >> 3;           // 0..31  (row for p-th pass: r0g + 32p)
    const int c4  = (tid & 7) * 4;      // 0..28

    v8f acc[2][4];
#pragma unroll
    for (int i = 0; i < 2; ++i)
#pragma unroll
        for (int j = 0; j < 4; ++j) acc[i][j] = (v8f){};

    // stage one 128x32 chunk pair into buffer `buf`
    auto stage = [&](int buf, int k0) {
#pragma unroll
        for (int p = 0; p < 4; ++p) {
            int r = r0g + 32 * p;
            async_cp16(&As[buf][r][c4], q + (size_t)(R0 + r) * 512 + k0 + c4);
            async_cp16(&Bs[buf][r][c4], q + (size_t)(C0 + r) * 512 + k0 + c4);
        }
    };

    stage(0, 0);
    wait_async0();
    __syncthreads();                     // buffer 0 ready

    int cur = 0;
    for (int k0 = 0; k0 < 512; k0 += KC) {
        if (k0 + KC < 512) stage(cur ^ 1, k0 + KC);   // prefetch next chunk

        // A frag (16x4 f32): lanes 0-15 hold K=k,k+1 ; lanes 16-31 K=k+2,k+3.
        // B frag (4x16 f32): same per-lane float2 pattern, N = lane&15.
#pragma unroll
        for (int kk = 0; kk < KC; kk += 4) {
            int kl = kk + 2 * half;
            v2f a0 = *(const v2f*)&As[cur][rowOff + l16][kl];
            v2f a1 = *(const v2f*)&As[cur][rowOff + 16 + l16][kl];
            v2f b0 = *(const v2f*)&Bs[cur][colOff + l16][kl];
            v2f b1 = *(const v2f*)&Bs[cur][colOff + 16 + l16][kl];
            v2f b2 = *(const v2f*)&Bs[cur][colOff + 32 + l16][kl];
            v2f b3 = *(const v2f*)&Bs[cur][colOff + 48 + l16][kl];
            acc[0][0] = __builtin_amdgcn_wmma_f32_16x16x4_f32(false, a0, false, b0, (short)0, acc[0][0], false, false);
            acc[0][1] = __builtin_amdgcn_wmma_f32_16x16x4_f32(false, a0, false, b1, (short)0, acc[0][1], false, false);
            acc[0][2] = __builtin_amdgcn_wmma_f32_16x16x4_f32(false, a0, false, b2, (short)0, acc[0][2], false, false);
            acc[0][3] = __builtin_amdgcn_wmma_f32_16x16x4_f32(false, a0, false, b3, (short)0, acc[0][3], false, false);
            acc[1][0] = __builtin_amdgcn_wmma_f32_16x16x4_f32(false, a1, false, b0, (short)0, acc[1][0], false, false);
            acc[1][1] = __builtin_amdgcn_wmma_f32_16x16x4_f32(false, a1, false, b1, (short)0, acc[1][1], false, false);
            acc[1][2] = __builtin_amdgcn_wmma_f32_16x16x4_f32(false, a1, false, b2, (short)0, acc[1][2], false, false);
            acc[1][3] = __builtin_amdgcn_wmma_f32_16x16x4_f32(false, a1, false, b3, (short)0, acc[1][3], false, false);
        }

        wait_async0();                   // own prefetch copies landed in LDS
        __syncthreads();                 // all waves done reading + writing
        cur ^= 1;
    }

    // Fused epilogue: C/D layout -> VGPR e holds M = e + 8*half, N = lane&15.
    float local = 0.0f;
#pragma unroll
    for (int i = 0; i < 2; ++i) {
#pragma unroll
        for (int e = 0; e < 8; ++e) {
            int   m   = R0 + rowOff + 16 * i + e + 8 * half;
            float snm = sqn[m];
#pragma unroll
            for (int j = 0; j < 4; ++j) {
                int   n  = C0 + colOff + 16 * j + l16;
                float g  = acc[i][j][e];
                float sq = fmaxf(snm + sqn[n] - 2.0f * g, 0.0f);
                float d  = (sq > 0.0f) ? sqrtf(sq) : 0.0f;
                float sm = expf(-d);
                local += sm * sm;
            }
        }
    }
    red[tid] = local;
    __syncthreads();
    for (int off = 128; off > 0; off >>= 1) {
        if (tid < off) red[tid] += red[tid + off];
        __syncthreads();
    }
    if (tid == 0) {
        float wgt = (bx == by) ? 1.0f : 2.0f;    // mirror block contribution
        partial[pid] = wgt * red[0];
    }
}

// ---------------------------------------------------------------------------
// Kernel 4: deterministic finalize: mean NLL + sqrt(sum of penalty partials)
// ---------------------------------------------------------------------------
__global__ void finalize_kernel(const float* __restrict__ nll,
                                const int* __restrict__ label,
                                const float* __restrict__ partial,
                                int N, int P, float* __restrict__ out) {
    __shared__ float rce[256];
    __shared__ float rpn[256];
    __shared__ int   rvc[256];
    float ce = 0.0f, pen = 0.0f;
    int valid = 0;
    for (int i = threadIdx.x; i < N; i += 256) {
        ce += nll[i];
        valid += (label[i] != IGNORE_INDEX) ? 1 : 0;
    }
    for (int i = threadIdx.x; i < P; i += 256) pen += partial[i];
    rce[threadIdx.x] = ce;
    rpn[threadIdx.x] = pen;
    rvc[threadIdx.x] = valid;
    __syncthreads();
    for (int off = 128; off > 0; off >>= 1) {
        if (threadIdx.x < off) {
            rce[threadIdx.x] += rce[threadIdx.x + off];
            rpn[threadIdx.x] += rpn[threadIdx.x + off];
            rvc[threadIdx.x] += rvc[threadIdx.x + off];
        }
        __syncthreads();
    }
    if (threadIdx.x == 0) {
        int nv = rvc[0] > 0 ? rvc[0] : 1;
        out[0] = rce[0] / (float)nv + sqrtf(rpn[0]);
    }
}

// ---------------------------------------------------------------------------
extern "C" void kernel_launch(void* const* d_in, const int* in_sizes, int n_in,
                              void* d_out, int out_size, void* d_ws, size_t ws_size,
                              hipStream_t stream) {
    const float* input = (const float*)d_in[0];   // [N, V] fp32
    const int*   label = (const int*)d_in[1];     // [N] int32
    const float* query = (const float*)d_in[2];   // [Q, 512] fp32

    const int N  = in_sizes[1];            // 8192
    const int V  = in_sizes[0] / N;        // 32000
    const int Q  = in_sizes[2] / 512;      // 8192
    const int GT = Q / BT;                 // 64 tiles per dim
    const int P  = GT * GT;                // 4096 penalty partial slots

    float* ws      = (float*)d_ws;
    float* nll     = ws;                   // [N]
    float* sqn     = ws + N;               // [Q]
    float* partial = ws + N + Q;           // [P]

    qsqnorm_kernel<<<Q, 128, 0, stream>>>(query, sqn);
    ce_rows_kernel<<<N, 256, 0, stream>>>(input, label, nll, V);
    gram_penalty_kernel<<<dim3(GT, GT), 256, 0, stream>>>(query, sqn, partial);
    finalize_kernel<<<1, 256, 0, stream>>>(nll, label, partial, N, P, (float*)d_out);
}